// RamanTwoPeaks_57337813401907
// MI455X (gfx1250) — compile-verified
//
#include <hip/hip_runtime.h>

// RamanTwoPeaks: elementwise two-Lorentzian + linear baseline over f32[65536,1024].
// HBM-bound (512 MB traffic, ~1 GFLOP): optimum is a wave32 streaming kernel.
// CDNA5 paths used: async global->LDS staging (ASYNCcnt pipeline), NT store
// temporal hints (preserve x in the 192MB L2 across replays), v_rcp_f32.

typedef float v4f __attribute__((ext_vector_type(4)));
typedef int   v4i __attribute__((vector_size(16)));   // matches builtin's pointee type

#define NROWS 4           // rows per block (software pipeline depth)
#define TPB   256         // 8 wave32 waves; 256 lanes * 4 floats = 1024 = XLEN

__device__ __forceinline__ float fast_rcp(float v) {
  return __builtin_amdgcn_rcpf(v);   // v_rcp_f32, ~1 ulp — fine vs f32 divide
}

__device__ __forceinline__ v4f raman_eval(v4f xv,
                                          float a0, float c0, float is0,
                                          float a1, float c1, float is1,
                                          float sl, float ct) {
  v4f ov;
#pragma unroll
  for (int i = 0; i < 4; ++i) {
    float x  = xv[i];
    float u0 = (x - c0) * is0;
    float u1 = (x - c1) * is1;
    float p0 = a0 * fast_rcp(__builtin_fmaf(u0, u0, 1.0f));
    float p1 = a1 * fast_rcp(__builtin_fmaf(u1, u1, 1.0f));
    ov[i] = (__builtin_fmaf(sl, x, ct) + p0) + p1;
  }
  return ov;
}

#if __has_builtin(__builtin_amdgcn_global_load_async_to_lds_b128)
#define HAVE_ASYNC_LDS 1
#else
#define HAVE_ASYNC_LDS 0
#endif

#if HAVE_ASYNC_LDS
// Fast path: requires xlen == 4*TPB and pixels % NROWS == 0 (true for the
// reference shapes). Each thread async-stages its own 16B per row into LDS,
// then drains rows in issue order with staggered s_wait_asynccnt so compute
// on row r overlaps the in-flight loads of rows r+1..3.
__global__ __launch_bounds__(TPB) void raman_async_kernel(
    const float* __restrict__ x,
    const float* __restrict__ a0p, const float* __restrict__ c0p,
    const float* __restrict__ s0p,
    const float* __restrict__ a1p, const float* __restrict__ c1p,
    const float* __restrict__ s1p,
    const float* __restrict__ slp, const float* __restrict__ ctp,
    float* __restrict__ out, int xlen) {
  __shared__ float tile[NROWS * TPB * 4];   // 16 KB staging buffer

  const int tid  = threadIdx.x;
  const int row0 = blockIdx.x * NROWS;
  const int col  = tid * 4;

  // Issue all NROWS async b128 loads up front (ASYNCcnt climbs to 4).
#pragma unroll
  for (int r = 0; r < NROWS; ++r) {
    const float* gsrc = x + (size_t)(row0 + r) * xlen + col;
    float*       ldst = &tile[(r * TPB + tid) * 4];
    __builtin_amdgcn_global_load_async_to_lds_b128(
        (__attribute__((address_space(1))) v4i*)(unsigned long long)(const void*)gsrc,
        (__attribute__((address_space(3))) v4i*)ldst,
        /*offset=*/0, /*cpol=*/0);
  }

#define PROCESS_ROW(r, cnt)                                                   \
  {                                                                           \
    asm volatile("s_wait_asynccnt " #cnt ::: "memory");                       \
    const int row = row0 + (r);                                               \
    const float A0 = a0p[row], C0 = c0p[row];                                 \
    const float IS0 = fast_rcp(s0p[row]);                                     \
    const float A1 = a1p[row], C1 = c1p[row];                                 \
    const float IS1 = fast_rcp(s1p[row]);                                     \
    const float SL = slp[row], CT = ctp[row];                                 \
    v4f xv = *(const v4f*)&tile[((r) * TPB + tid) * 4];                       \
    v4f ov = raman_eval(xv, A0, C0, IS0, A1, C1, IS1, SL, CT);                \
    __builtin_nontemporal_store(ov, (v4f*)(out + (size_t)row * xlen + col));  \
  }

  PROCESS_ROW(0, 3)
  PROCESS_ROW(1, 2)
  PROCESS_ROW(2, 1)
  PROCESS_ROW(3, 0)
#undef PROCESS_ROW
}
#endif  // HAVE_ASYNC_LDS

// Generic / fallback path: direct b128 loads + global_prefetch of the next
// row, NT b128 stores. Handles arbitrary shapes.
__global__ __launch_bounds__(TPB) void raman_direct_kernel(
    const float* __restrict__ x,
    const float* __restrict__ a0p, const float* __restrict__ c0p,
    const float* __restrict__ s0p,
    const float* __restrict__ a1p, const float* __restrict__ c1p,
    const float* __restrict__ s1p,
    const float* __restrict__ slp, const float* __restrict__ ctp,
    float* __restrict__ out, int pixels, int xlen) {
  const int tid = threadIdx.x;
#pragma unroll
  for (int r = 0; r < NROWS; ++r) {
    const int row = blockIdx.x * NROWS + r;
    if (row >= pixels) return;
    const float A0 = a0p[row], C0 = c0p[row];
    const float IS0 = fast_rcp(s0p[row]);
    const float A1 = a1p[row], C1 = c1p[row];
    const float IS1 = fast_rcp(s1p[row]);
    const float SL = slp[row], CT = ctp[row];

    const float* xrow = x + (size_t)row * xlen;
    float*       orow = out + (size_t)row * xlen;

    for (int c = tid * 4; c + 3 < xlen; c += TPB * 4) {
      if (row + 1 < pixels)
        __builtin_prefetch(xrow + xlen + c, 0, 3);  // global_prefetch_b8
      v4f xv = *(const v4f*)(xrow + c);
      v4f ov = raman_eval(xv, A0, C0, IS0, A1, C1, IS1, SL, CT);
      __builtin_nontemporal_store(ov, (v4f*)(orow + c));
    }
    // scalar tail (xlen % 4 leftovers)
    for (int c = (xlen & ~3) + tid; c < xlen; c += TPB) {
      float xs = xrow[c];
      float u0 = (xs - C0) * IS0;
      float u1 = (xs - C1) * IS1;
      float p0 = A0 * fast_rcp(__builtin_fmaf(u0, u0, 1.0f));
      float p1 = A1 * fast_rcp(__builtin_fmaf(u1, u1, 1.0f));
      orow[c] = (__builtin_fmaf(SL, xs, CT) + p0) + p1;
    }
  }
}

extern "C" void kernel_launch(void* const* d_in, const int* in_sizes, int n_in,
                              void* d_out, int out_size, void* d_ws, size_t ws_size,
                              hipStream_t stream) {
  const float* x  = (const float*)d_in[0];
  const float* a0 = (const float*)d_in[1];
  const float* c0 = (const float*)d_in[2];
  const float* s0 = (const float*)d_in[3];
  const float* a1 = (const float*)d_in[4];
  const float* c1 = (const float*)d_in[5];
  const float* s1 = (const float*)d_in[6];
  const float* sl = (const float*)d_in[7];
  const float* ct = (const float*)d_in[8];
  float* out = (float*)d_out;

  const int pixels = in_sizes[1];                     // param arrays are [pixels]
  const int xlen   = in_sizes[0] / (pixels > 0 ? pixels : 1);
  const int nblocks = (pixels + NROWS - 1) / NROWS;

#if HAVE_ASYNC_LDS
  if (xlen == TPB * 4 && (pixels % NROWS) == 0) {
    raman_async_kernel<<<nblocks, TPB, 0, stream>>>(
        x, a0, c0, s0, a1, c1, s1, sl, ct, out, xlen);
    return;
  }
#endif
  raman_direct_kernel<<<nblocks, TPB, 0, stream>>>(
      x, a0, c0, s0, a1, c1, s1, sl, ct, out, pixels, xlen);
}